// FlexConvolution_45251775430799
// MI455X (gfx1250) — compile-verified
//
#include <hip/hip_runtime.h>

typedef __attribute__((ext_vector_type(2))) float v2f;
typedef __attribute__((ext_vector_type(4))) float v4f;
typedef __attribute__((ext_vector_type(8))) float v8f;

#define NPTS 32768
#define NB   4
#define CIN  64
#define COUT 64
#define KNB  16

// W' stored pair-interleaved: addr = dcp*WPAIRSTR + o*2 + (dc&1)
// WPAIRSTR = 160 dwords == 32 (mod 64)  -> lanes 16-31 (dcp+1) hit the other bank half
#define WPAIRSTR 160
#define WSZ  (128 * WPAIRSTR)            // 20480 floats
#define VOFF WSZ
#define VWAVE (128 * 32)                 // v tile pair-interleaved: addr = dcp*32 + nl*2 + p
#define BOFF (VOFF + 4 * VWAVE)          // bias (64 floats)
#define LDS_FLOATS (BOFF + 64)           // 36928 floats = ~144 KB

// ---------------- features transpose: [B,Cin,N] -> [B,N,Cin] in d_ws ----------------
__global__ __launch_bounds__(256) void transpose_feat_kernel(
    const float* __restrict__ feat, float* __restrict__ ft)
{
    __shared__ float tile[64][65];
    const int b  = blockIdx.y;
    const int n0 = blockIdx.x * 64;
    const int tid = threadIdx.x;
    #pragma unroll
    for (int i = 0; i < 16; ++i) {
        const int j = tid + i * 256;
        const int c = j >> 6, nloc = j & 63;
        tile[nloc][c] = feat[(b * CIN + c) * NPTS + n0 + nloc];   // coalesced along n
    }
    __syncthreads();
    #pragma unroll
    for (int i = 0; i < 16; ++i) {
        const int j = tid + i * 256;
        const int nloc = j >> 6, c = j & 63;
        ft[(b * NPTS + n0 + nloc) * CIN + c] = tile[nloc][c];     // coalesced along c
    }
}

// ---------------------------------- main kernel ----------------------------------
template <bool USE_FT>
__global__ __launch_bounds__(128) void flexconv_wmma_kernel(
    const float* __restrict__ feat,   // [B, Cin, N]
    const float* __restrict__ ft,     // [B, N, Cin] (transposed, if USE_FT)
    const float* __restrict__ wt,     // [3, Cin, Cout]  == W'[0:192][:]
    const float* __restrict__ wb,     // [Cin, Cout]     == W'[192:256][:]
    const float* __restrict__ bias,   // [Cout]
    const int*   __restrict__ nbr,    // [B, K, N]
    const float* __restrict__ pos,    // [B, 3, N]
    float* __restrict__ out)          // [B, Cout, N]
{
    __shared__ float lds[LDS_FLOATS];
    const int tid = threadIdx.x;

    // ---- stage W' pair-interleaved + bias into LDS ----
    for (int j = tid; j < 128 * 64; j += 128) {      // j = dcp*64 + o
        const int dcp = j >> 6, o = j & 63;
        const int dc = dcp * 2;
        float w0, w1;
        if (dc < 192) { w0 = wt[dc * 64 + o];        w1 = wt[(dc + 1) * 64 + o]; }
        else          { w0 = wb[(dc - 192) * 64 + o]; w1 = wb[(dc - 191) * 64 + o]; }
        v2f p; p.x = w0; p.y = w1;
        *reinterpret_cast<v2f*>(&lds[dcp * WPAIRSTR + o * 2]) = p;
    }
    if (tid < COUT) lds[BOFF + tid] = bias[tid];
    __syncthreads();

    const int wave = tid >> 5;
    const int lane = tid & 31;
    const int nl   = lane & 15;   // point within the wave's 16-point tile
    const int h    = lane >> 4;   // half-lane group (channel half / K-slot half)
    const int b    = blockIdx.y;
    const int n    = blockIdx.x * 64 + wave * 16 + nl;
    const int vb   = VOFF + wave * VWAVE;

    // ================= phase 1: neighbor contraction -> v[256][16] ============
    const float cx = pos[(b * 3 + 0) * NPTS + n];
    const float cy = pos[(b * 3 + 1) * NPTS + n];
    const float cz = pos[(b * 3 + 2) * NPTS + n];

    int   idxs[KNB];
    float dxs[KNB], dys[KNB], dzs[KNB];
    #pragma unroll
    for (int k = 0; k < KNB; ++k)
        idxs[k] = nbr[(b * KNB + k) * NPTS + n];
    #pragma unroll
    for (int k = 0; k < KNB; ++k) {
        const int id = idxs[k];
        dxs[k] = pos[(b * 3 + 0) * NPTS + id] - cx;
        dys[k] = pos[(b * 3 + 1) * NPTS + id] - cy;
        dzs[k] = pos[(b * 3 + 2) * NPTS + id] - cz;
    }

    for (int cc = 0; cc < 4; ++cc) {                 // 4 chunks of 8 channels
        const int cbase = h * 32 + cc * 8;
        float ax[8] = {}, ay[8] = {}, az[8] = {}, asum[8] = {};
        for (int k = 0; k < KNB; ++k) {
            const int   id = idxs[k];
            const float dx = dxs[k], dy = dys[k], dz = dzs[k];
            float f[8];
            if constexpr (USE_FT) {
                const v4f* fp = reinterpret_cast<const v4f*>(ft + (b * NPTS + id) * CIN + cbase);
                const v4f f0 = fp[0], f1 = fp[1];    // two contiguous b128 gathers
                #pragma unroll
                for (int c = 0; c < 4; ++c) { f[c] = f0[c]; f[c + 4] = f1[c]; }
            } else {
                const float* fp = feat + (b * CIN + cbase) * NPTS + id;
                #pragma unroll
                for (int c = 0; c < 8; ++c) f[c] = fp[c * NPTS];
            }
            #pragma unroll
            for (int c = 0; c < 8; ++c) {
                ax[c]   = fmaf(f[c], dx, ax[c]);
                ay[c]   = fmaf(f[c], dy, ay[c]);
                az[c]   = fmaf(f[c], dz, az[c]);
                asum[c] += f[c];
            }
        }
        // pair-interleaved b64 stores: v addr = (dc>>1)*32 + nl*2 + (dc&1)
        #pragma unroll
        for (int c = 0; c < 8; c += 2) {
            const int dcp = (cbase + c) >> 1;        // even base -> clean pair
            v2f px, py, pz, ps;
            px.x = ax[c];   px.y = ax[c + 1];
            py.x = ay[c];   py.y = ay[c + 1];
            pz.x = az[c];   pz.y = az[c + 1];
            ps.x = asum[c]; ps.y = asum[c + 1];
            *reinterpret_cast<v2f*>(&lds[vb + (dcp      ) * 32 + nl * 2]) = px; // d=0
            *reinterpret_cast<v2f*>(&lds[vb + (dcp + 32 ) * 32 + nl * 2]) = py; // d=1
            *reinterpret_cast<v2f*>(&lds[vb + (dcp + 64 ) * 32 + nl * 2]) = pz; // d=2
            *reinterpret_cast<v2f*>(&lds[vb + (dcp + 96 ) * 32 + nl * 2]) = ps; // d=3
        }
    }
    // v tile is wave-private; same-wave LDS ops are in-order (DScnt), no barrier

    // ==== phase 2: out[64 x 16] = W'^T[64 x 256] * v[256 x 16] via WMMA f32 ====
    // A (16x4, MxK): lanes hold M=o; VGPR0 = {K0 | K2}, VGPR1 = {K1 | K3}
    // B (4x16, KxN): lanes hold N=pt; VGPR0 = {K0 | K2}, VGPR1 = {K1 | K3}
    v8f acc[4] = {v8f{}, v8f{}, v8f{}, v8f{}};
    #pragma unroll 4
    for (int ks = 0; ks < 64; ++ks) {
        const int dcp = ks * 2 + h;                  // this lane's K-slot pair
        const v2f bf = *reinterpret_cast<const v2f*>(&lds[vb + dcp * 32 + nl * 2]);
        #pragma unroll
        for (int ot = 0; ot < 4; ++ot) {
            const v2f af = *reinterpret_cast<const v2f*>(
                &lds[dcp * WPAIRSTR + (ot * 16 + nl) * 2]);
            acc[ot] = __builtin_amdgcn_wmma_f32_16x16x4_f32(
                false, af, false, bf, (short)0, acc[ot], false, false);
        }
    }

    // ---- epilogue: C/D layout -> o = ot*16 + r + h*8, n = tile base + nl ----
    #pragma unroll
    for (int ot = 0; ot < 4; ++ot) {
        #pragma unroll
        for (int r = 0; r < 8; ++r) {
            const int o = ot * 16 + r + h * 8;
            out[(b * COUT + o) * NPTS + n] = acc[ot][r] + lds[BOFF + o];
        }
    }
}

extern "C" void kernel_launch(void* const* d_in, const int* in_sizes, int n_in,
                              void* d_out, int out_size, void* d_ws, size_t ws_size,
                              hipStream_t stream) {
    (void)in_sizes; (void)n_in; (void)out_size;
    const float* feat = (const float*)d_in[0];   // features     [B, Cin, N] f32
    const float* wt   = (const float*)d_in[1];   // weight_theta [3, Cin, Cout] f32
    const float* wb   = (const float*)d_in[2];   // weight_bias  [Cin, Cout] f32
    const float* bias = (const float*)d_in[3];   // bias [Cout] f32
    const int*   nbr  = (const int*)d_in[4];     // neighborhood [B, K, N] i32
    const float* pos  = (const float*)d_in[5];   // positions [B, 3, N] f32
    float* out = (float*)d_out;                  // [B, Cout, N] f32

    const size_t ft_bytes = (size_t)NB * NPTS * CIN * sizeof(float);  // 32 MB
    dim3 grid(NPTS / 64, NB);

    if (ws_size >= ft_bytes) {
        float* ft = (float*)d_ws;
        transpose_feat_kernel<<<grid, 256, 0, stream>>>(feat, ft);
        flexconv_wmma_kernel<true><<<grid, 128, 0, stream>>>(
            feat, ft, wt, wb, bias, nbr, pos, out);
    } else {
        flexconv_wmma_kernel<false><<<grid, 128, 0, stream>>>(
            feat, feat, wt, wb, bias, nbr, pos, out);
    }
}